// ScaledDotProductAttention_77601469104403
// MI455X (gfx1250) — compile-verified
//
#include <hip/hip_runtime.h>

typedef __attribute__((ext_vector_type(2))) float v2f;
typedef __attribute__((ext_vector_type(4))) float v4f;
typedef __attribute__((ext_vector_type(8))) float v8f;

#define NTOK 2048
#define DK   64
#define RS   2052          // padded LDS row stride (floats): 2052 % 64 == 4 -> conflict-free
#define NEGV -1e9f

// One block = 256 threads = 8 wave32, handles a 16-row query stripe of one batch.
// Full 16 x 2048 score stripe lives in LDS (CDNA5: 320KB/WGP makes this possible).
__global__ __launch_bounds__(256) void attn_fused_kernel(
    const float* __restrict__ Q, const float* __restrict__ K,
    const float* __restrict__ V,
    const unsigned char* __restrict__ KPM,   // (N, M) bool
    const unsigned char* __restrict__ AM,    // (M, M) bool
    float* __restrict__ OUT,                 // (N, M, DK)
    float* __restrict__ ATT)                 // (N, M, M)
{
    extern __shared__ float smem[];
    float* P     = smem;                 // 16 * RS
    float* pvscr = smem + 16 * RS;       // 4 * 256

    const int tid    = threadIdx.x;
    const int wave   = tid >> 5;           // wave32
    const int lane   = tid & 31;
    const int lg     = lane >> 4;          // lane group (K-half select in WMMA layout)
    const int ll     = lane & 15;          // M/N index within fragment
    const int mblock = blockIdx.x;
    const int batch  = blockIdx.y;

    const float* Qb = Q + (size_t)batch * NTOK * DK;
    const float* Kb = K + (size_t)batch * NTOK * DK;
    const float* Vb = V + (size_t)batch * NTOK * DK;
    const unsigned char* kpmb = KPM + (size_t)batch * NTOK;

    // ---- preload Q tile A-fragments into registers (reused for this wave's 16 col tiles) ----
    // A 16x4 f32 layout: lane L holds row m=L%16; VGPR0=K(2*lg), VGPR1=K(2*lg+1)
    v2f aQ[16];
    {
        const float* qrow = Qb + (size_t)(mblock * 16 + ll) * DK;
        #pragma unroll
        for (int dc = 0; dc < 16; ++dc)
            aQ[dc] = *(const v2f*)(qrow + dc * 4 + lg * 2);
    }

    // ---- phase 1: S = (Q K^T)/8 + masks, for this wave's 256-column slice ----
    const int colbase = wave * 256;
    for (int t = 0; t < 16; ++t) {
        const int n0 = colbase + t * 16;
        v8f acc = {};
        const float* krow = Kb + (size_t)(n0 + ll) * DK + lg * 2;  // B[kb][n] = K[n][kb]
        #pragma unroll
        for (int dc = 0; dc < 16; ++dc) {
            v2f b = *(const v2f*)(krow + dc * 4);
            acc = __builtin_amdgcn_wmma_f32_16x16x4_f32(
                false, aQ[dc], false, b, (short)0, acc, false, false);
        }
        // D layout: element vv -> row m = vv + 8*lg, col n = n0 + ll
        const int col  = n0 + ll;
        const float ka = kpmb[col] ? NEGV : 0.0f;
        #pragma unroll
        for (int vv = 0; vv < 8; ++vv) {
            const int m = vv + 8 * lg;
            const float aa = AM[(size_t)(mblock * 16 + m) * NTOK + col] ? NEGV : 0.0f;
            P[m * RS + col] = acc[vv] * 0.125f + ka + aa;
        }
    }
    __syncthreads();

    // ---- phase 2: row softmax. thread (r, seg) handles float4s at c = j*64 + seg*4
    // (conflict-free LDS banks; 16 threads of a row cover a contiguous 256B global run)
    const int r    = tid >> 4;
    const int seg  = tid & 15;
    float* prow = P + r * RS;
    const int cbase = seg * 4;

    float lm = -3.4e38f;
    #pragma unroll 8
    for (int j = 0; j < 32; ++j) {
        v4f x = *(const v4f*)(prow + j * 64 + cbase);
        lm = fmaxf(lm, fmaxf(fmaxf(x.x, x.y), fmaxf(x.z, x.w)));
    }
    // row max: the 16 segments of a row sit in one 16-lane half of this wave32
    #pragma unroll
    for (int off = 1; off < 16; off <<= 1)
        lm = fmaxf(lm, __shfl_xor(lm, off, 32));
    const float rowmax = lm;

    float ls = 0.0f;
    #pragma unroll 4
    for (int j = 0; j < 32; ++j) {
        v4f x = *(const v4f*)(prow + j * 64 + cbase);
        x.x = __expf(x.x - rowmax);
        x.y = __expf(x.y - rowmax);
        x.z = __expf(x.z - rowmax);
        x.w = __expf(x.w - rowmax);
        *(v4f*)(prow + j * 64 + cbase) = x;
        ls += (x.x + x.y) + (x.z + x.w);
    }
    #pragma unroll
    for (int off = 1; off < 16; off <<= 1)
        ls += __shfl_xor(ls, off, 32);
    const float inv = 1.0f / ls;

    float* arow = ATT + (size_t)batch * NTOK * NTOK
                      + (size_t)(mblock * 16 + r) * NTOK;
    #pragma unroll 4
    for (int j = 0; j < 32; ++j) {
        const int c = j * 64 + cbase;
        v4f x = *(const v4f*)(prow + c);
        x *= inv;
        *(v4f*)(prow + c) = x;                              // normalized P for phase 3
        __builtin_nontemporal_store(x, (v4f*)(arow + c));   // NT: don't evict K/V from L2
    }
    __syncthreads();

    // ---- phase 3: OUT = P @ V (4 d-tiles x 2 column-halves across 8 waves) ----
    const int dt   = wave & 3;
    const int half = wave >> 2;
    const int d0   = dt * 16;
    v8f accO = {};
    const float* vcol = Vb + d0 + ll;
    const float* pArow = P + ll * RS;    // A fragment: row m = ll
    for (int n = half * 1024; n < half * 1024 + 1024; n += 4) {
        const int kk = n + lg * 2;
        v2f a = *(const v2f*)(pArow + kk);
        v2f b;
        b.x = vcol[(size_t)kk * DK];
        b.y = vcol[(size_t)(kk + 1) * DK];
        accO = __builtin_amdgcn_wmma_f32_16x16x4_f32(
            false, a, false, b, (short)0, accO, false, false);
    }
    if (half == 1) {
        #pragma unroll
        for (int vv = 0; vv < 8; ++vv)
            pvscr[dt * 256 + lane * 8 + vv] = accO[vv];
    }
    __syncthreads();
    if (half == 0) {
        float* orow = OUT + (size_t)batch * NTOK * DK + (size_t)(mblock * 16) * DK;
        #pragma unroll
        for (int vv = 0; vv < 8; ++vv) {
            const int m = vv + 8 * lg;
            orow[(size_t)m * DK + d0 + ll] = accO[vv] + pvscr[dt * 256 + lane * 8 + vv];
        }
    }
}

extern "C" void kernel_launch(void* const* d_in, const int* in_sizes, int n_in,
                              void* d_out, int out_size, void* d_ws, size_t ws_size,
                              hipStream_t stream) {
    (void)in_sizes; (void)n_in; (void)out_size; (void)d_ws; (void)ws_size;
    const float*         Qp  = (const float*)d_in[0];
    const float*         Kp  = (const float*)d_in[1];
    const float*         Vp  = (const float*)d_in[2];
    const unsigned char* KPM = (const unsigned char*)d_in[3];
    const unsigned char* AM  = (const unsigned char*)d_in[4];

    float* out = (float*)d_out;                        // (32, 2048, 64) first
    float* att = out + (size_t)32 * 2048 * 64;         // then (32, 2048, 2048)

    const size_t shmem = (size_t)(16 * RS + 4 * 256) * sizeof(float);
    dim3 grid(NTOK / 16, 32);   // 128 row-stripes x 32 batches
    attn_fused_kernel<<<grid, 256, shmem, stream>>>(Qp, Kp, Vp, KPM, AM, out, att);
}